// GCN_86122684219473
// MI455X (gfx1250) — compile-verified
//
#include <hip/hip_runtime.h>

// ---------------------------------------------------------------------------
// GCN forward for MI455X (gfx1250, wave32, WMMA).
//   support = x @ W1                     (WMMA f32 16x16x4)
//   h       = relu(spmm(adj, support) + b1)
//   out2    = relu(h) @ W2               (WMMA f32 16x16x4, relu fused in load)
//   out     = spmm(adj, out2) + b2
// Biases folded into buffer initialization. SPMM = scatter with fp32 atomics
// (all intermediates fit in the 192MB L2, so atomics are L2-resident).
// Weights staged in LDS interleaved by K-pairs so each WMMA B-fragment is one
// aligned ds_load_b64 (no register repacking movs in the inner loop).
// ---------------------------------------------------------------------------

typedef __attribute__((ext_vector_type(2))) float v2f;
typedef __attribute__((ext_vector_type(4))) float v4f;
typedef __attribute__((ext_vector_type(8))) float v8f;

#define GCN_NODES 100000
#define GCN_EDGES 1600000
#define GCN_F 128
#define GCN_H 64
#define GCN_O 32

// ---- init: h = b1 broadcast, out = b2 broadcast -----------------------------
__global__ void __launch_bounds__(256) gcn_init(float* __restrict__ h,
                                                float* __restrict__ out,
                                                const float* __restrict__ b1,
                                                const float* __restrict__ b2) {
  long i = (long)blockIdx.x * 256 + threadIdx.x;        // 0 .. 6.4M-1
  if (i < (long)GCN_NODES * GCN_H) h[i] = b1[i & (GCN_H - 1)];
  if (i < (long)GCN_NODES * GCN_O) out[i] = b2[i & (GCN_O - 1)];
}

// ---- GEMM1: support[100000x64] = x[100000x128] @ W1[128x64] ----------------
// Block = 256 thr = 8 waves; block covers 128 rows; wave -> 16x64 strip
// (4 accumulators of 16x16). W1 staged in LDS (32KB), K-pair interleaved:
//   sW[((k>>1)*64 + n)*2 + (k&1)] = W1[k*64 + n]
// so B fragment {W1[ka][n], W1[ka+1][n]} is one contiguous 8B ds_load_b64.
__global__ void __launch_bounds__(256) gcn_gemm1(const float* __restrict__ x,
                                                 const float* __restrict__ W1,
                                                 float* __restrict__ sup) {
  __shared__ float sW[GCN_F * GCN_H];                   // 8192 f32 = 32KB
  const int tid = threadIdx.x;
  for (int i = tid; i < GCN_F * GCN_H; i += 256) {
    const int k = i >> 6;                               // i / 64
    const int n = i & 63;
    sW[(((k >> 1) * GCN_H) + n) * 2 + (k & 1)] = W1[i];
  }
  __syncthreads();

  const int lane = tid & 31;
  const int wave = tid >> 5;
  const int half = lane >> 4;                           // 0: K pair k..k+1, 1: k+2..k+3
  const int l16  = lane & 15;
  const int m0   = blockIdx.x * 128 + wave * 16;

  int mrow = m0 + l16;
  if (mrow >= GCN_NODES) mrow = GCN_NODES - 1;          // clamp (stores guarded)
  const float* __restrict__ xrow = x + (long)mrow * GCN_F;

  v8f acc0 = {}, acc1 = {}, acc2 = {}, acc3 = {};
  for (int k = 0; k < GCN_F; k += 4) {
    const int ka = k + 2 * half;                        // this lane's K pair (even)
    v2f a = *(const v2f*)&xrow[ka];                     // A frag: M=l16, K=ka,ka+1
    const v2f* bp = (const v2f*)&sW[(((ka >> 1) * GCN_H) + l16) * 2];
    acc0 = __builtin_amdgcn_wmma_f32_16x16x4_f32(false, a, false, bp[0],  (short)0, acc0, false, false);
    acc1 = __builtin_amdgcn_wmma_f32_16x16x4_f32(false, a, false, bp[16], (short)0, acc1, false, false);
    acc2 = __builtin_amdgcn_wmma_f32_16x16x4_f32(false, a, false, bp[32], (short)0, acc2, false, false);
    acc3 = __builtin_amdgcn_wmma_f32_16x16x4_f32(false, a, false, bp[48], (short)0, acc3, false, false);
  }

  // C/D layout: VGPR v -> M = v + 8*half, N = l16
  const int rbase = m0 + half * 8;
  for (int v = 0; v < 8; ++v) {
    const int r = rbase + v;
    if (r < GCN_NODES) {
      float* o = sup + (long)r * GCN_H + l16;
      o[0] = acc0[v]; o[16] = acc1[v]; o[32] = acc2[v]; o[48] = acc3[v];
    }
  }
}

// ---- SPMM1: h[dst] += val * support[src], D = 64 ---------------------------
// One wave per edge; each lane owns a float2 of the feature row (coalesced).
__global__ void __launch_bounds__(256) gcn_spmm1(const float* __restrict__ sup,
                                                 float* __restrict__ h,
                                                 const int* __restrict__ esrc,
                                                 const int* __restrict__ edst,
                                                 const float* __restrict__ vals) {
  const int lane = threadIdx.x & 31;
  const long e = (long)blockIdx.x * 8 + (threadIdx.x >> 5);
  if (e >= GCN_EDGES) return;
  const int s = esrc[e];
  const int d = edst[e];
  const float v = vals[e];
  v2f g = *(const v2f*)&sup[(long)s * GCN_H + 2 * lane];
  float* hp = &h[(long)d * GCN_H + 2 * lane];
  unsafeAtomicAdd(hp,     g.x * v);                     // global_atomic_add_f32
  unsafeAtomicAdd(hp + 1, g.y * v);
}

// ---- GEMM2: out2[100000x32] = relu(h)[100000x64] @ W2[64x32] ---------------
__global__ void __launch_bounds__(256) gcn_gemm2(const float* __restrict__ h,
                                                 const float* __restrict__ W2,
                                                 float* __restrict__ o2) {
  __shared__ float sW[GCN_H * GCN_O];                   // 2048 f32 = 8KB
  const int tid = threadIdx.x;
  for (int i = tid; i < GCN_H * GCN_O; i += 256) {
    const int k = i >> 5;                               // i / 32
    const int n = i & 31;
    sW[(((k >> 1) * GCN_O) + n) * 2 + (k & 1)] = W2[i];
  }
  __syncthreads();

  const int lane = tid & 31;
  const int wave = tid >> 5;
  const int half = lane >> 4;
  const int l16  = lane & 15;
  const int m0   = blockIdx.x * 128 + wave * 16;

  int mrow = m0 + l16;
  if (mrow >= GCN_NODES) mrow = GCN_NODES - 1;
  const float* __restrict__ hrow = h + (long)mrow * GCN_H;

  v8f acc0 = {}, acc1 = {};
  for (int k = 0; k < GCN_H; k += 4) {
    const int ka = k + 2 * half;
    v2f a = *(const v2f*)&hrow[ka];
    a.x = fmaxf(a.x, 0.0f);                             // fused relu
    a.y = fmaxf(a.y, 0.0f);
    const v2f* bp = (const v2f*)&sW[(((ka >> 1) * GCN_O) + l16) * 2];
    acc0 = __builtin_amdgcn_wmma_f32_16x16x4_f32(false, a, false, bp[0],  (short)0, acc0, false, false);
    acc1 = __builtin_amdgcn_wmma_f32_16x16x4_f32(false, a, false, bp[16], (short)0, acc1, false, false);
  }

  const int rbase = m0 + half * 8;
  for (int v = 0; v < 8; ++v) {
    const int r = rbase + v;
    if (r < GCN_NODES) {
      float* o = o2 + (long)r * GCN_O + l16;
      o[0] = acc0[v]; o[16] = acc1[v];
    }
  }
}

// ---- SPMM2: out[dst] += val * out2[src], D = 32 ----------------------------
// Two edges per wave: lanes 0-15 -> edge e, lanes 16-31 -> edge e+1.
__global__ void __launch_bounds__(256) gcn_spmm2(const float* __restrict__ o2,
                                                 float* __restrict__ out,
                                                 const int* __restrict__ esrc,
                                                 const int* __restrict__ edst,
                                                 const float* __restrict__ vals) {
  const int lane = threadIdx.x & 31;
  const int sub  = lane >> 4;                           // edge within wave
  const int f    = (lane & 15) * 2;                     // feature pair
  const long e = (long)blockIdx.x * 16 + (threadIdx.x >> 5) * 2 + sub;
  if (e >= GCN_EDGES) return;
  const int s = esrc[e];
  const int d = edst[e];
  const float v = vals[e];
  v2f g = *(const v2f*)&o2[(long)s * GCN_O + f];
  float* op = &out[(long)d * GCN_O + f];
  unsafeAtomicAdd(op,     g.x * v);
  unsafeAtomicAdd(op + 1, g.y * v);
}

// ---------------------------------------------------------------------------
extern "C" void kernel_launch(void* const* d_in, const int* in_sizes, int n_in,
                              void* d_out, int out_size, void* d_ws, size_t ws_size,
                              hipStream_t stream) {
  const float* x    = (const float*)d_in[0];
  const float* adjv = (const float*)d_in[1];
  const float* W1   = (const float*)d_in[2];
  const float* b1   = (const float*)d_in[3];
  const float* W2   = (const float*)d_in[4];
  const float* b2   = (const float*)d_in[5];
  const int*   esrc = (const int*)d_in[6];
  const int*   edst = (const int*)d_in[7];
  float* out = (float*)d_out;

  // Workspace layout (floats): support | h | out2  = 16M f32 = 64MB
  float* ws  = (float*)d_ws;
  float* sup = ws;                                   // 100000*64
  float* h   = ws + (long)GCN_NODES * GCN_H;         // 100000*64
  float* o2  = ws + (long)GCN_NODES * GCN_H * 2;     // 100000*32

  const int rowBlocks = (GCN_NODES + 127) / 128;     // 782

  gcn_init <<<(GCN_NODES * GCN_H + 255) / 256, 256, 0, stream>>>(h, out, b1, b2);
  gcn_gemm1<<<rowBlocks, 256, 0, stream>>>(x, W1, sup);
  gcn_spmm1<<<GCN_EDGES / 8, 256, 0, stream>>>(sup, h, esrc, edst, adjv);
  gcn_gemm2<<<rowBlocks, 256, 0, stream>>>(h, W2, o2);
  gcn_spmm2<<<GCN_EDGES / 16, 256, 0, stream>>>(o2, out, esrc, edst, adjv);
}